// HouseGCN_26319559590755
// MI455X (gfx1250) — compile-verified
//
#include <hip/hip_runtime.h>

// ---------------------------------------------------------------------------
// 2-layer GCN for MI455X (gfx1250, wave32).
// Dense GEMMs use V_WMMA_F32_16X16X4_F32 (full fp32 precision, matrix pipe).
// Edge aggregation is HBM-bound: float4 gathers + global_atomic_add_f32.
// Self-loop term folded into the epilogue: out[i] += h[i]*dis[i]^2 + bias.
// ---------------------------------------------------------------------------

typedef __attribute__((ext_vector_type(2))) float v2f;
typedef __attribute__((ext_vector_type(8))) float v8f;

#define N_NODES 100000

// ---------------- utility fills ----------------
__global__ void fill_kernel(float* __restrict__ p, float v, int n) {
    int i = blockIdx.x * blockDim.x + threadIdx.x;
    if (i < n) p[i] = v;
}

// deg[col[e]] += ew[e]   (deg pre-filled with 1.0 for the self-loop weight)
__global__ void deg_kernel(const int* __restrict__ col, const float* __restrict__ ew,
                           float* __restrict__ deg, int E) {
    int e = blockIdx.x * blockDim.x + threadIdx.x;
    if (e < E) atomicAdd(&deg[col[e]], ew[e]);
}

// dis = deg > 0 ? rsqrt(deg) : 0   (in place)
__global__ void rsqrt_kernel(float* __restrict__ d, int n) {
    int i = blockIdx.x * blockDim.x + threadIdx.x;
    if (i < n) {
        float v = d[i];
        d[i] = (v > 0.f) ? rsqrtf(v) : 0.f;
    }
}

// ---------------- WMMA fp32 GEMM: H[M,NOUT] = X[M,K] @ W[NOUT,K]^T ----------
// One wave per 16x16 output tile. blockDim = (32, NOUT/16); blockIdx.x = M/16.
// A fragment: lane l<16 -> row m0+l, K={k,k+1}; lanes 16-31 -> K={k+2,k+3}.
// B fragment mirrors A with W rows as columns of W^T.
// D: VGPR v of lane l -> (m0 + v + 8*(l>=16), n0 + (l&15)).
template <int K, int NOUT>
__global__ void gemm_wmma_kernel(const float* __restrict__ X,
                                 const float* __restrict__ W,
                                 float* __restrict__ H) {
    const int lane = threadIdx.x;        // 0..31
    const int half = lane >> 4;          // 0 or 1
    const int l    = lane & 15;
    const int m0   = blockIdx.x << 4;
    const int n0   = threadIdx.y << 4;

    const float* arow = X + (m0 + l) * K + (half << 1);
    const float* brow = W + (n0 + l) * K + (half << 1);

    v8f acc = {};
#pragma unroll
    for (int k = 0; k < K; k += 4) {
        v2f a = *(const v2f*)(arow + k);
        v2f b = *(const v2f*)(brow + k);
        // D = A x B + C  (v_wmma_f32_16x16x4_f32)
        acc = __builtin_amdgcn_wmma_f32_16x16x4_f32(
            /*neg_a=*/false, a, /*neg_b=*/false, b,
            /*c_mod=*/(short)0, acc, /*reuse_a=*/false, /*reuse_b=*/false);
    }

    float* out = H + (m0 + (half << 3)) * NOUT + n0 + l;
#pragma unroll
    for (int v = 0; v < 8; ++v) out[v * NOUT] = acc[v];
}

// ---------------- edge scatter: out[col] += h[row] * (dis[r]*ew*dis[c]) -----
// FEAT/4 threads per edge, each handles 4 contiguous features (float4 gather).
template <int FEAT>
__global__ void scatter_kernel(const float* __restrict__ H,
                               const int* __restrict__ row,
                               const int* __restrict__ col,
                               const float* __restrict__ ew,
                               const float* __restrict__ dis,
                               float* __restrict__ out, int E) {
    constexpr int G = FEAT / 4;
    int tid = blockIdx.x * blockDim.x + threadIdx.x;
    int e = tid / G;
    if (e >= E) return;
    int g = (tid - e * G) << 2;

    int r = row[e];
    int c = col[e];
    float nrm = dis[r] * ew[e] * dis[c];

    float4 v = *(const float4*)(H + r * FEAT + g);
    float* dst = out + c * FEAT + g;
    atomicAdd(dst + 0, v.x * nrm);
    atomicAdd(dst + 1, v.y * nrm);
    atomicAdd(dst + 2, v.z * nrm);
    atomicAdd(dst + 3, v.w * nrm);
}

// ---------------- epilogue: out += h*dis^2 + bias, optional ReLU ------------
template <int FEAT, bool RELU>
__global__ void finish_kernel(const float* __restrict__ H,
                              const float* __restrict__ dis,
                              const float* __restrict__ bias,
                              float* __restrict__ out, int n4) {
    constexpr int G = FEAT / 4;
    int i4 = blockIdx.x * blockDim.x + threadIdx.x;
    if (i4 >= n4) return;
    int node = i4 / G;
    int f = (i4 - node * G) << 2;

    float ds = dis[node];
    float d2 = ds * ds;

    float4 h = *(const float4*)(H + node * FEAT + f);
    float4 b = *(const float4*)(bias + f);
    float4 o = *(const float4*)(out + node * FEAT + f);

    float4 r;
    r.x = fmaf(h.x, d2, o.x) + b.x;
    r.y = fmaf(h.y, d2, o.y) + b.y;
    r.z = fmaf(h.z, d2, o.z) + b.z;
    r.w = fmaf(h.w, d2, o.w) + b.w;
    if (RELU) {
        r.x = fmaxf(r.x, 0.f);
        r.y = fmaxf(r.y, 0.f);
        r.z = fmaxf(r.z, 0.f);
        r.w = fmaxf(r.w, 0.f);
    }
    *(float4*)(out + node * FEAT + f) = r;
}

// ---------------------------------------------------------------------------
extern "C" void kernel_launch(void* const* d_in, const int* in_sizes, int n_in,
                              void* d_out, int out_size, void* d_ws, size_t ws_size,
                              hipStream_t stream) {
    const float* x  = (const float*)d_in[0];
    const int*   ei = (const int*)d_in[1];   // [2, E] flat
    const float* ew = (const float*)d_in[2];
    const float* W1 = (const float*)d_in[3]; // [64,128]
    const float* b1 = (const float*)d_in[4]; // [64]
    const float* W2 = (const float*)d_in[5]; // [32,64]
    const float* b2 = (const float*)d_in[6]; // [32]
    float* out = (float*)d_out;              // [N,32]

    const int E = in_sizes[2];
    const int* row = ei;
    const int* col = ei + E;

    // workspace layout (floats):
    //   dis : N            (degree, then rsqrt in place)
    //   h1  : N*64         (x @ W1^T; later reused as h2 = a1 @ W2^T)
    //   a1  : N*64         (aggregated layer-1 activations)
    float* ws  = (float*)d_ws;
    float* dis = ws;
    float* h1  = ws + N_NODES;
    float* a1  = h1 + (size_t)N_NODES * 64;
    float* h2  = h1;  // h1 dead after layer-1 epilogue

    const int T = 256;
    const int MT = N_NODES / 16;  // 6250 exact

    // ---- normalization precompute ----
    fill_kernel<<<(N_NODES + T - 1) / T, T, 0, stream>>>(dis, 1.0f, N_NODES); // self-loop wt
    fill_kernel<<<((N_NODES * 64) + T - 1) / T, T, 0, stream>>>(a1, 0.0f, N_NODES * 64);
    fill_kernel<<<((N_NODES * 32) + T - 1) / T, T, 0, stream>>>(out, 0.0f, N_NODES * 32);
    deg_kernel<<<(E + T - 1) / T, T, 0, stream>>>(col, ew, dis, E);
    rsqrt_kernel<<<(N_NODES + T - 1) / T, T, 0, stream>>>(dis, N_NODES);

    // ---- layer 1: h1 = x @ W1^T (WMMA fp32) ----
    gemm_wmma_kernel<128, 64><<<MT, dim3(32, 4), 0, stream>>>(x, W1, h1);

    // ---- layer 1: edge aggregation + self-loop/bias/ReLU ----
    {
        int nthr = E * 16;  // 16 threads per edge (64 feats / 4)
        scatter_kernel<64><<<(nthr + T - 1) / T, T, 0, stream>>>(h1, row, col, ew, dis, a1, E);
        int n4 = N_NODES * 16;
        finish_kernel<64, true><<<(n4 + T - 1) / T, T, 0, stream>>>(h1, dis, b1, a1, n4);
    }

    // ---- layer 2: h2 = a1 @ W2^T (WMMA fp32) ----
    gemm_wmma_kernel<64, 32><<<MT, dim3(32, 2), 0, stream>>>(a1, W2, h2);

    // ---- layer 2: edge aggregation + self-loop/bias ----
    {
        int nthr = E * 8;   // 8 threads per edge (32 feats / 4)
        scatter_kernel<32><<<(nthr + T - 1) / T, T, 0, stream>>>(h2, row, col, ew, dis, out, E);
        int n4 = N_NODES * 8;
        finish_kernel<32, false><<<(n4 + T - 1) / T, T, 0, stream>>>(h2, dis, b2, out, n4);
    }
}